// DomainShiftRegularizer_25451976196743
// MI455X (gfx1250) — compile-verified
//
#include <hip/hip_runtime.h>
#include <hip/hip_bf16.h>

// ---------------------------------------------------------------------------
// Sizes from the reference
// ---------------------------------------------------------------------------
#define NN      1024          // N = B*NODES
#define CC      768
#define NPROJ_  50
#define NPAD    64            // projection count padded to 4 N-tiles
#define NT      100           // STEPS * NPROJ
#define HBINS   4096
#define PAIRS_F 1048576.0f    // 1024*1024 per kernel block

typedef __attribute__((ext_vector_type(2))) float v2f;
typedef __attribute__((ext_vector_type(8))) float v8f;

// ---------------------------------------------------------------------------
// 0) zero a u32 range (hist + per-t state + sums)
// ---------------------------------------------------------------------------
__global__ void dsr_zero_u32(unsigned* __restrict__ p, int n) {
  int i = blockIdx.x * 256 + threadIdx.x;
  if (i < n) p[i] = 0u;
}

// ---------------------------------------------------------------------------
// 1) normalize projection rows into a ZERO-PADDED 64x768 buffer:
//    rows 0..49 = proj / max(||row||,1e-12); rows 50..63 = 0.
//    Padding removes all conditionals from the WMMA inner loop.
// ---------------------------------------------------------------------------
__global__ __launch_bounds__(256) void dsr_norm_dirs(const float* __restrict__ proj,
                                                     float* __restrict__ dirs) {
  __shared__ float red[256];
  const int r = blockIdx.x, tid = threadIdx.x;
  if (r >= NPROJ_) {
#pragma unroll
    for (int i = 0; i < 3; ++i) dirs[r * CC + tid + i * 256] = 0.0f;
    return;
  }
  float s = 0.f;
#pragma unroll
  for (int i = 0; i < 3; ++i) { float v = proj[r * CC + tid + i * 256]; s += v * v; }
  red[tid] = s; __syncthreads();
  for (int st = 128; st > 0; st >>= 1) { if (tid < st) red[tid] += red[tid + st]; __syncthreads(); }
  const float sc = 1.0f / fmaxf(sqrtf(red[0]), 1e-12f);
#pragma unroll
  for (int i = 0; i < 3; ++i) { int c = tid + i * 256; dirs[r * CC + c] = proj[r * CC + c] * sc; }
}

// ---------------------------------------------------------------------------
// 2) WMMA FP32 GEMM: X(1024 x 64) = A(1024 x 768) * dirsPad^T(768 x 64)
//    One wave per 16x16 tile; V_WMMA_F32_16X16X4_F32, K-loop step 4.
//    A layout (32b A 16x4): lanes 0-15 hold K=0,1; lanes 16-31 hold K=2,3.
//    B layout mirrors A (4x16).  C/D: VGPR r -> M=r (lanes 0-15), M=r+8 (16-31).
//    Inner loop: 2 x global_load_b64 + 1 x v_wmma, no EXEC games.
// ---------------------------------------------------------------------------
__global__ __launch_bounds__(32) void dsr_wmma_gemm(const float* __restrict__ A,
                                                    const float* __restrict__ Dm,
                                                    float* __restrict__ X) {
  const int tile = blockIdx.x;        // 0..255
  const int tm = tile >> 2;           // 64 M-tiles
  const int tn = tile & 3;            // 4 N-tiles
  const int lane = threadIdx.x & 31;
  const int l15 = lane & 15;
  const int hi = lane >> 4;           // 0 or 1
  const float* __restrict__ ap = A + (tm * 16 + l15) * CC + (hi << 1);
  const float* __restrict__ bp = Dm + (tn * 16 + l15) * CC + (hi << 1);
  v8f acc = {};
#pragma unroll 8
  for (int k = 0; k < CC; k += 4) {
    const v2f a = *(const v2f*)(ap + k);
    const v2f b = *(const v2f*)(bp + k);
    acc = __builtin_amdgcn_wmma_f32_16x16x4_f32(false, a, false, b,
                                                (short)0, acc, false, false);
  }
  const int rbase = tm * 16 + (hi << 3);
#pragma unroll
  for (int r = 0; r < 8; ++r)
    X[(rbase + r) * NPAD + tn * 16 + l15] = acc[r];
}

// ---------------------------------------------------------------------------
// 3) normalize: blocks 0..49 -> x columns (1/(||col||+1e-8)),
//               blocks 50..51 -> y rows from batch_y (STEPS,B,NODES transpose)
// ---------------------------------------------------------------------------
__global__ __launch_bounds__(256) void dsr_normalize_xy(float* __restrict__ xall,
                                                        const float* __restrict__ by,
                                                        float* __restrict__ ynorm) {
  __shared__ float red[256];
  const int b = blockIdx.x, tid = threadIdx.x;
  if (b < NPROJ_) {
    float s = 0.f;
#pragma unroll
    for (int i = 0; i < 4; ++i) { float v = xall[(tid + i * 256) * NPAD + b]; s += v * v; }
    red[tid] = s; __syncthreads();
    for (int st = 128; st > 0; st >>= 1) { if (tid < st) red[tid] += red[tid + st]; __syncthreads(); }
    const float sc = 1.0f / (sqrtf(red[0]) + 1e-8f);
#pragma unroll
    for (int i = 0; i < 4; ++i) { int n = tid + i * 256; xall[n * NPAD + b] *= sc; }
  } else {
    const int sI = b - NPROJ_;
    float s = 0.f;
#pragma unroll
    for (int i = 0; i < 4; ++i) {
      int n = tid + i * 256;
      float v = by[(n >> 6) * 128 + sI * 64 + (n & 63)];
      s += v * v;
    }
    red[tid] = s; __syncthreads();
    for (int st = 128; st > 0; st >>= 1) { if (tid < st) red[tid] += red[tid + st]; __syncthreads(); }
    const float sc = 1.0f / (sqrtf(red[0]) + 1e-8f);
#pragma unroll
    for (int i = 0; i < 4; ++i) {
      int n = tid + i * 256;
      ynorm[sI * NN + n] = by[(n >> 6) * 128 + sI * 64 + (n & 63)] * sc;
    }
  }
}

// ---------------------------------------------------------------------------
// 4) per-t point arrays: px = x col i, py = y row s, pz = y row s permuted
// ---------------------------------------------------------------------------
__global__ __launch_bounds__(256) void dsr_build_pts(const float* __restrict__ xall,
                                                     const float* __restrict__ ynorm,
                                                     const int* __restrict__ perms,
                                                     float* __restrict__ pts) {
  const int t = blockIdx.x, tid = threadIdx.x;
  const int sI = t / NPROJ_, i = t % NPROJ_;
  float* p = pts + t * 3 * NN;
#pragma unroll
  for (int c = 0; c < 4; ++c) {
    int n = tid + c * 256;
    p[n]          = xall[n * NPAD + i];
    p[NN + n]     = ynorm[sI * NN + n];
    p[2 * NN + n] = ynorm[sI * NN + perms[t * NN + n]];
  }
}

// ---------------------------------------------------------------------------
// 5) radix-select histogram pass over the three 1024x1024 distance blocks.
//    grid = (8 row-slabs, 3 block types, 100 t).  XY counted with weight 2
//    (the full 2048^2 matrix contains XY and YX).  Keys = IEEE bits of d^2
//    (order-preserving for non-negative floats); mode 1/2/3 = bits 31:20 /
//    19:8 / 7:0.
// ---------------------------------------------------------------------------
__global__ __launch_bounds__(256) void dsr_hist_pass(const float* __restrict__ pts,
                                                     const unsigned* __restrict__ st_prefix,
                                                     unsigned* __restrict__ hist,
                                                     int mode) {
  __shared__ unsigned sh[HBINS];
  __shared__ float sbx[NN], sby[NN], sax[128], say[128];
  const int t = blockIdx.z, bt = blockIdx.y;
  const int r0 = blockIdx.x * 128;
  const int tid = threadIdx.x;
  const float* px = pts + t * 3 * NN;
  const float* py = px + NN;
  const float* pz = px + 2 * NN;
  const float* ay_p = (bt == 1) ? pz : py;   // rows: joint unless YY
  const float* by_p = (bt == 0) ? py : pz;   // cols: joint only for XX
  const unsigned w = (bt == 2) ? 2u : 1u;
  const unsigned pref = st_prefix[t];
#pragma unroll
  for (int i = 0; i < HBINS / 256; ++i) sh[tid + i * 256] = 0u;
#pragma unroll
  for (int i = 0; i < 4; ++i) { int j = tid + i * 256; sbx[j] = px[j]; sby[j] = by_p[j]; }
  if (tid < 128) { sax[tid] = px[r0 + tid]; say[tid] = ay_p[r0 + tid]; }
  __syncthreads();
  for (int r = 0; r < 128; ++r) {
    const float ax = sax[r], ay = say[r];
#pragma unroll
    for (int c = 0; c < 4; ++c) {
      const int j = tid + c * 256;
      const float dx = ax - sbx[j], dy = ay - sby[j];
      const float d2 = dx * dx + dy * dy;
      if (d2 > 0.0f) {
        const unsigned key = __float_as_uint(d2);
        if (mode == 1) {
          atomicAdd(&sh[key >> 20], w);
        } else if (mode == 2) {
          if ((key >> 20) == (pref >> 20)) atomicAdd(&sh[(key >> 8) & 0xFFFu], w);
        } else {
          if ((key >> 8) == (pref >> 8)) atomicAdd(&sh[key & 0xFFu], w);
        }
      }
    }
  }
  __syncthreads();
  unsigned* gh = hist + (size_t)t * HBINS;
#pragma unroll
  for (int i = 0; i < HBINS / 256; ++i) {
    const int b = tid + i * 256;
    const unsigned v = sh[b];
    if (v) atomicAdd(&gh[b], v);
  }
}

// ---------------------------------------------------------------------------
// 6) per-t rank selection across the histogram; re-zeros hist for next pass.
//    mode 1: m = #positives, k = (m-1)/2 (torch lower median); mode 3 emits
//    gamma = 1/sqrt(selected d^2).
// ---------------------------------------------------------------------------
__global__ __launch_bounds__(256) void dsr_scan_pass(unsigned* __restrict__ hist,
                                                     unsigned* __restrict__ st_prefix,
                                                     unsigned* __restrict__ st_krem,
                                                     unsigned* __restrict__ st_m,
                                                     float* __restrict__ st_gamma,
                                                     int mode) {
  __shared__ unsigned part[256];
  const int t = blockIdx.x, tid = threadIdx.x;
  unsigned* gh = hist + (size_t)t * HBINS;
  const int nbins = (mode == 3) ? 256 : HBINS;
  const int per = nbins / 256;
  unsigned ps = 0;
  for (int i = 0; i < per; ++i) ps += gh[tid * per + i];
  part[tid] = ps;
  __syncthreads();
  if (tid == 0) {
    unsigned k;
    if (mode == 1) {
      unsigned m = 0;
      for (int i = 0; i < 256; ++i) m += part[i];
      st_m[t] = m;
      k = (m > 0) ? ((m - 1) >> 1) : 0;
    } else {
      k = st_krem[t];
    }
    unsigned cum = 0; int chunk = 0;
    for (int i = 0; i < 256; ++i) {
      if (cum + part[i] > k) { chunk = i; break; }
      cum += part[i];
    }
    int b = chunk * per;
    for (int i = 0; i < per; ++i) {
      const unsigned c = gh[chunk * per + i];
      if (cum + c > k) { b = chunk * per + i; break; }
      cum += c;
    }
    const unsigned krem = k - cum;
    if (mode == 1)      { st_prefix[t] = ((unsigned)b) << 20; st_krem[t] = krem; }
    else if (mode == 2) { st_prefix[t] |= ((unsigned)b) << 8; st_krem[t] = krem; }
    else {
      const unsigned key = st_prefix[t] | (unsigned)b;
      const float med = sqrtf(__uint_as_float(key));
      st_gamma[t] = (st_m[t] > 0 && med > 0.0f) ? (1.0f / med) : 1.0f;
    }
  }
  __syncthreads();
#pragma unroll
  for (int i = 0; i < HBINS / 256; ++i) gh[tid + i * 256] = 0u;
}

// ---------------------------------------------------------------------------
// 7) RBF kernel sums: Sxx, Syy, Sxy per t (means -> finalize)
// ---------------------------------------------------------------------------
__global__ __launch_bounds__(256) void dsr_mmd_sum(const float* __restrict__ pts,
                                                   const float* __restrict__ st_gamma,
                                                   float* __restrict__ sums) {
  __shared__ float sbx[NN], sby[NN], sax[128], say[128], red[256];
  const int t = blockIdx.z, bt = blockIdx.y;
  const int r0 = blockIdx.x * 128;
  const int tid = threadIdx.x;
  const float* px = pts + t * 3 * NN;
  const float* py = px + NN;
  const float* pz = px + 2 * NN;
  const float* ay_p = (bt == 1) ? pz : py;
  const float* by_p = (bt == 0) ? py : pz;
  const float g = st_gamma[t];
#pragma unroll
  for (int i = 0; i < 4; ++i) { int j = tid + i * 256; sbx[j] = px[j]; sby[j] = by_p[j]; }
  if (tid < 128) { sax[tid] = px[r0 + tid]; say[tid] = ay_p[r0 + tid]; }
  __syncthreads();
  float acc = 0.f;
  for (int r = 0; r < 128; ++r) {
    const float ax = sax[r], ay = say[r];
#pragma unroll
    for (int c = 0; c < 4; ++c) {
      const int j = tid + c * 256;
      const float dx = ax - sbx[j], dy = ay - sby[j];
      acc += __expf(-g * (dx * dx + dy * dy));
    }
  }
  red[tid] = acc; __syncthreads();
  for (int st = 128; st > 0; st >>= 1) { if (tid < st) red[tid] += red[tid + st]; __syncthreads(); }
  if (tid == 0) atomicAdd(&sums[t * 3 + bt], red[0]);
}

// ---------------------------------------------------------------------------
// 8) finalize: mmd_t = sqrt(max((Sxx+Syy-2Sxy)/1M, 0)); reg = mean over 100
// ---------------------------------------------------------------------------
__global__ __launch_bounds__(128) void dsr_finalize(const float* __restrict__ sums,
                                                    const float* __restrict__ base,
                                                    float* __restrict__ out) {
  __shared__ float red[128];
  const int tid = threadIdx.x;
  float v = 0.f;
  if (tid < NT) {
    const float Sxx = sums[tid * 3 + 0], Syy = sums[tid * 3 + 1], Sxy = sums[tid * 3 + 2];
    const float mmd2 = (Sxx + Syy - 2.0f * Sxy) * (1.0f / PAIRS_F);
    v = sqrtf(fmaxf(mmd2, 0.0f));
  }
  red[tid] = v; __syncthreads();
  for (int s = 64; s > 0; s >>= 1) { if (tid < s) red[tid] += red[tid + s]; __syncthreads(); }
  if (tid == 0) {
    const float reg = red[0] / (float)NT;
    out[0] = base[0] + 0.01f * reg;
    out[1] = reg;
  }
}

// ---------------------------------------------------------------------------
extern "C" void kernel_launch(void* const* d_in, const int* in_sizes, int n_in,
                              void* d_out, int out_size, void* d_ws, size_t ws_size,
                              hipStream_t stream) {
  const float* plm   = (const float*)d_in[0];   // (16,64,768)
  const float* by    = (const float*)d_in[1];   // (16,2,64)
  const float* base  = (const float*)d_in[2];   // scalar
  const float* proj  = (const float*)d_in[3];   // (50,768)
  const int*   perms = (const int*)d_in[4];     // (100,1024)
  float* out = (float*)d_out;

  // workspace layout
  float* ws_f = (float*)d_ws;
  float* dirs  = ws_f;                          // 49152 f (64x768, zero-padded)
  float* xall  = dirs + NPAD * CC;              // 65536 f (1024 x 64)
  float* ynorm = xall + NN * NPAD;              // 2048 f
  float* pts   = ynorm + 2 * NN;                // 307200 f
  unsigned* hist      = (unsigned*)(pts + NT * 3 * NN);   // 409600 u32
  unsigned* st_prefix = hist + NT * HBINS;                // 100
  unsigned* st_krem   = st_prefix + NT;                   // 100
  unsigned* st_m      = st_krem + NT;                     // 100
  float*    st_gamma  = (float*)(st_m + NT);              // 100
  float*    sums      = st_gamma + NT;                    // 300

  const int nzero = NT * HBINS + 3 * NT + NT + 3 * NT;    // hist + state + gamma + sums
  dsr_zero_u32<<<(nzero + 255) / 256, 256, 0, stream>>>(hist, nzero);

  dsr_norm_dirs<<<NPAD, 256, 0, stream>>>(proj, dirs);
  dsr_wmma_gemm<<<256, 32, 0, stream>>>(plm, dirs, xall);
  dsr_normalize_xy<<<NPROJ_ + 2, 256, 0, stream>>>(xall, by, ynorm);
  dsr_build_pts<<<NT, 256, 0, stream>>>(xall, ynorm, perms, pts);

  const dim3 hgrid(8, 3, NT);
  dsr_hist_pass<<<hgrid, 256, 0, stream>>>(pts, st_prefix, hist, 1);
  dsr_scan_pass<<<NT, 256, 0, stream>>>(hist, st_prefix, st_krem, st_m, st_gamma, 1);
  dsr_hist_pass<<<hgrid, 256, 0, stream>>>(pts, st_prefix, hist, 2);
  dsr_scan_pass<<<NT, 256, 0, stream>>>(hist, st_prefix, st_krem, st_m, st_gamma, 2);
  dsr_hist_pass<<<hgrid, 256, 0, stream>>>(pts, st_prefix, hist, 3);
  dsr_scan_pass<<<NT, 256, 0, stream>>>(hist, st_prefix, st_krem, st_m, st_gamma, 3);

  dsr_mmd_sum<<<hgrid, 256, 0, stream>>>(pts, st_gamma, sums);
  dsr_finalize<<<1, 128, 0, stream>>>(sums, base, out);
}